// PaiNNInteraction_16535624090180
// MI455X (gfx1250) — compile-verified
//
#include <hip/hip_runtime.h>
#include <hip/hip_bf16.h>

typedef __attribute__((ext_vector_type(2))) float v2f;
typedef __attribute__((ext_vector_type(4))) float f4;
typedef __attribute__((ext_vector_type(8))) float v8f;

#define FDIM 128
#define F3   384
#define LDS_STRIDE 132   // 132 mod 64 == 4 word-banks/row -> conflict-free b64 A-frag reads

// ---------------------------------------------------------------------------
// Kernel 1: context MLP  cfea = (silu(s@W1+b1))@W2 + b2   via v_wmma_f32_16x16x4_f32
// block = 256 threads (8 waves), handles 64 atoms (4 row-tiles of 16)
// ---------------------------------------------------------------------------
__global__ __launch_bounds__(256) void painn_mlp_wmma(
    const float* __restrict__ s,
    const float* __restrict__ W1, const float* __restrict__ b1,
    const float* __restrict__ W2, const float* __restrict__ b2,
    float* __restrict__ cfea, int n_atoms)
{
    __shared__ float tile[64 * LDS_STRIDE];

    const int tid     = threadIdx.x;
    const int lane    = tid & 31;
    const int wv      = tid >> 5;          // 0..7  -> n-tile id
    const int rowbase = blockIdx.x * 64;

    const int mrow  = lane & 15;           // A row within 16-tile / C col
    const int khalf = lane >> 4;           // 0/1 -> K sub-pair

    // ---- stage 0: load 64x128 s-tile into LDS (zero-pad OOB rows) ----
    #pragma unroll
    for (int it = 0; it < 8; ++it) {
        int fid  = it * 256 + tid;         // float4 id, 2048 total (32 per row)
        int row  = fid >> 5;
        int c4   = fid & 31;
        f4 v = {0.f, 0.f, 0.f, 0.f};
        int grow = rowbase + row;
        if (grow < n_atoms) v = *(const f4*)(s + (size_t)grow * FDIM + c4 * 4);
        *(f4*)(&tile[row * LDS_STRIDE + c4 * 4]) = v;
    }
    __syncthreads();

    const int ncol = wv * 16 + mrow;       // h column in [0,128)
    const float bias1 = b1[ncol];

    // ---- stage 1: h_pre = s @ W1 ----
    v8f acc[4];
    #pragma unroll
    for (int rt = 0; rt < 4; ++rt) acc[rt] = (v8f){0.f,0.f,0.f,0.f,0.f,0.f,0.f,0.f};

    for (int k0 = 0; k0 < FDIM; k0 += 4) {
        const int k = k0 + khalf * 2;
        v2f b;
        b.x = W1[(size_t)k * FDIM + ncol];
        b.y = W1[(size_t)(k + 1) * FDIM + ncol];
        #pragma unroll
        for (int rt = 0; rt < 4; ++rt) {
            v2f a = *(const v2f*)(&tile[(rt * 16 + mrow) * LDS_STRIDE + k]);
            acc[rt] = __builtin_amdgcn_wmma_f32_16x16x4_f32(
                false, a, false, b, (short)0, acc[rt], false, false);
        }
    }

    // ---- silu + stash h into LDS (reuse same buffer) ----
    __syncthreads();
    #pragma unroll
    for (int rt = 0; rt < 4; ++rt) {
        #pragma unroll
        for (int r = 0; r < 8; ++r) {
            float x = acc[rt][r] + bias1;
            x = x / (1.0f + __expf(-x));   // silu
            tile[(rt * 16 + khalf * 8 + r) * LDS_STRIDE + ncol] = x;
        }
    }
    __syncthreads();

    // ---- stage 2: cfea = h @ W2 + b2 ; wave wv owns n-tiles {wv, wv+8, wv+16} ----
    v8f acc2[3][4];
    #pragma unroll
    for (int t = 0; t < 3; ++t)
        #pragma unroll
        for (int rt = 0; rt < 4; ++rt)
            acc2[t][rt] = (v8f){0.f,0.f,0.f,0.f,0.f,0.f,0.f,0.f};

    for (int k0 = 0; k0 < FDIM; k0 += 4) {
        const int k = k0 + khalf * 2;
        v2f bb[3];
        #pragma unroll
        for (int t = 0; t < 3; ++t) {
            int n = (wv + t * 8) * 16 + mrow;          // [0,384)
            bb[t].x = W2[(size_t)k * F3 + n];
            bb[t].y = W2[(size_t)(k + 1) * F3 + n];
        }
        #pragma unroll
        for (int rt = 0; rt < 4; ++rt) {
            v2f a = *(const v2f*)(&tile[(rt * 16 + mrow) * LDS_STRIDE + k]);
            #pragma unroll
            for (int t = 0; t < 3; ++t)
                acc2[t][rt] = __builtin_amdgcn_wmma_f32_16x16x4_f32(
                    false, a, false, bb[t], (short)0, acc2[t][rt], false, false);
        }
    }

    #pragma unroll
    for (int t = 0; t < 3; ++t) {
        const int n  = (wv + t * 8) * 16 + mrow;
        const float bias2 = b2[n];
        #pragma unroll
        for (int rt = 0; rt < 4; ++rt) {
            #pragma unroll
            for (int r = 0; r < 8; ++r) {
                int atom = rowbase + rt * 16 + khalf * 8 + r;
                if (atom < n_atoms)
                    cfea[(size_t)atom * F3 + n] = acc2[t][rt][r] + bias2;
            }
        }
    }
}

// ---------------------------------------------------------------------------
// Kernel 2: edge stage, deterministic segmented reduce (idx_i is sorted).
// One wave per atom; lane owns 4 features; binary-search edge range.
// descriptors are streamed once -> non-temporal loads keep cfea/efeat in L2.
// ---------------------------------------------------------------------------
__global__ __launch_bounds__(256) void painn_edge_reduce(
    const float* __restrict__ sfeat, const float* __restrict__ efeat,
    const float* __restrict__ desc,  const float* __restrict__ vecs,
    const int*   __restrict__ idx_i, const int*   __restrict__ idx_j,
    const float* __restrict__ cfea,  float* __restrict__ out,
    int n_atoms, int n_edges)
{
    const int i = blockIdx.x * 8 + (threadIdx.x >> 5);   // atom id (one wave each)
    if (i >= n_atoms) return;
    const int lane = threadIdx.x & 31;
    const int fb   = lane * 4;                           // 4 features per lane

    // lower_bound(idx_i, i) and lower_bound(idx_i, i+1)  (wave-uniform)
    int lo = 0, hi = n_edges;
    while (lo < hi) { int mid = (lo + hi) >> 1; if (idx_i[mid] < i) lo = mid + 1; else hi = mid; }
    const int start = lo;
    hi = n_edges;
    while (lo < hi) { int mid = (lo + hi) >> 1; if (idx_i[mid] < i + 1) lo = mid + 1; else hi = mid; }
    const int end = lo;

    f4 as  = {0.f,0.f,0.f,0.f};
    f4 ae0 = {0.f,0.f,0.f,0.f};
    f4 ae1 = {0.f,0.f,0.f,0.f};
    f4 ae2 = {0.f,0.f,0.f,0.f};

    for (int e = start; e < end; ++e) {
        const int j = idx_j[e];
        const size_t db = (size_t)e * F3 + fb;
        const size_t cb = (size_t)j * F3 + fb;
        // streamed-once descriptor tile: non-temporal
        const f4 d_s = __builtin_nontemporal_load((const f4*)(desc + db));
        const f4 d_v = __builtin_nontemporal_load((const f4*)(desc + db + 128));
        const f4 d_e = __builtin_nontemporal_load((const f4*)(desc + db + 256));
        const f4 c_s = *(const f4*)(cfea + cb);
        const f4 c_v = *(const f4*)(cfea + cb + 128);
        const f4 c_e = *(const f4*)(cfea + cb + 256);
        const float vx = vecs[(size_t)e * 3 + 0];
        const float vy = vecs[(size_t)e * 3 + 1];
        const float vz = vecs[(size_t)e * 3 + 2];
        const f4 f0 = *(const f4*)(efeat + ((size_t)j * 3 + 0) * FDIM + fb);
        const f4 f1 = *(const f4*)(efeat + ((size_t)j * 3 + 1) * FDIM + fb);
        const f4 f2 = *(const f4*)(efeat + ((size_t)j * 3 + 2) * FDIM + fb);

        as += d_s * c_s;
        const f4 mv = d_v * c_v;
        const f4 me = d_e * c_e;
        ae0 += mv * vx + me * f0;
        ae1 += mv * vy + me * f1;
        ae2 += mv * vz + me * f2;
    }

    // fused residual writes (deterministic, no atomics)
    *(f4*)(out + (size_t)i * FDIM + fb) = *(const f4*)(sfeat + (size_t)i * FDIM + fb) + as;
    float* oe = out + (size_t)n_atoms * FDIM;
    *(f4*)(oe + ((size_t)i * 3 + 0) * FDIM + fb) = *(const f4*)(efeat + ((size_t)i * 3 + 0) * FDIM + fb) + ae0;
    *(f4*)(oe + ((size_t)i * 3 + 1) * FDIM + fb) = *(const f4*)(efeat + ((size_t)i * 3 + 1) * FDIM + fb) + ae1;
    *(f4*)(oe + ((size_t)i * 3 + 2) * FDIM + fb) = *(const f4*)(efeat + ((size_t)i * 3 + 2) * FDIM + fb) + ae2;
}

// ---------------------------------------------------------------------------
extern "C" void kernel_launch(void* const* d_in, const int* in_sizes, int n_in,
                              void* d_out, int out_size, void* d_ws, size_t ws_size,
                              hipStream_t stream)
{
    const float* sfeat = (const float*)d_in[0];   // [n_atoms,1,128]
    const float* efeat = (const float*)d_in[1];   // [n_atoms,3,128]
    const float* desc  = (const float*)d_in[2];   // [n_edges,1,384]
    const float* vecs  = (const float*)d_in[3];   // [n_edges,3]
    const float* W1    = (const float*)d_in[4];   // [128,128]
    const float* b1    = (const float*)d_in[5];   // [128]
    const float* W2    = (const float*)d_in[6];   // [128,384]
    const float* b2    = (const float*)d_in[7];   // [384]
    const int*   idx_i = (const int*)d_in[8];     // [n_edges] sorted
    const int*   idx_j = (const int*)d_in[9];     // [n_edges]
    float*       out   = (float*)d_out;

    const int n_atoms = in_sizes[0] / FDIM;
    const int n_edges = in_sizes[8];

    float* cfea = (float*)d_ws;                   // n_atoms*384 floats (30.7 MB)

    dim3 blk(256);
    dim3 grd_mlp((n_atoms + 63) / 64);
    painn_mlp_wmma<<<grd_mlp, blk, 0, stream>>>(sfeat, W1, b1, W2, b2, cfea, n_atoms);

    dim3 grd_edge((n_atoms + 7) / 8);
    painn_edge_reduce<<<grd_edge, blk, 0, stream>>>(sfeat, efeat, desc, vecs,
                                                    idx_i, idx_j, cfea, out,
                                                    n_atoms, n_edges);
}